// SSIM_9457517986315
// MI455X (gfx1250) — compile-verified
//
#include <hip/hip_runtime.h>
#include <math.h>

// ---------------- types ----------------
typedef __attribute__((ext_vector_type(2))) float v2f;
typedef __attribute__((ext_vector_type(8))) float v8f;

#define IMGHW   512
#define OUTHW   502            // 512 - 11 + 1
#define TILE    16
#define HALO    26             // TILE + KSIZE - 1
#define PADK    28             // HALO padded to multiple of 4 (WMMA K chunks)
#define PSTR    28             // row stride (floats) for halo buffers
#define NTILES1D 32            // ceil(502/16)
#define C1_SSIM 0.0001f        // (0.01*1)^2
#define C2_SSIM 0.0009f        // (0.03*1)^2
#define N_TOTAL 24192384.0     // 32*3*502*502

// D = A(16x4) * B(4x16) + C, f32, wave32
__device__ __forceinline__ v8f wmma4(v2f a, v2f b, v8f c) {
    return __builtin_amdgcn_wmma_f32_16x16x4_f32(
        /*neg_a=*/false, a, /*neg_b=*/false, b,
        /*c_mod=*/(short)0, c, /*reuse_a=*/false, /*reuse_b=*/false);
}

// One wave (32 threads) per 16x16 output tile.
__global__ void __launch_bounds__(32)
ssim_tile_kernel(const float* __restrict__ X, const float* __restrict__ Y,
                 float* __restrict__ partial) {
    __shared__ __align__(16) float gk[16];            // normalized 1-D gaussian
    __shared__ __align__(16) float Xb[HALO * PSTR];
    __shared__ __align__(16) float Yb[HALO * PSTR];
    __shared__ __align__(16) float Pb[HALO * PSTR];   // current product map
    __shared__ __align__(16) float Hb[PADK * 16];     // horizontal-blur stage

    const int lane   = threadIdx.x;        // 0..31
    const int laneLo = lane & 15;
    const int laneHi = lane >> 4;          // 0 or 1

    const int b   = blockIdx.x;
    const int tx  = b & (NTILES1D - 1);
    const int ty  = (b >> 5) & (NTILES1D - 1);
    const int img = b >> 10;               // 0..95  (n*c)

    const float* xim = X + (size_t)img * (IMGHW * IMGHW);
    const float* yim = Y + (size_t)img * (IMGHW * IMGHW);

    // ---- build normalized 1-D gaussian in LDS (faithful to reference) ----
    if (lane < 16) {
        float v = 0.0f;
        if (lane < 11) {
            float t = ((float)lane - 5.0f) / 3.0f;   // (x-mean)/(2*sigma)
            v = expf(-t * t);
        }
        gk[lane] = v;
    }
    __syncthreads();
    float gsum = 0.0f;
    for (int i = 0; i < 11; ++i) gsum += gk[i];
    __syncthreads();
    if (lane < 11) gk[lane] = gk[lane] / gsum;
    __syncthreads();

    // ---- banded constant operand chunks: Bc[ch] lane -> gn[kk - (lane&15)] ----
    // Serves as B for the horizontal pass and as A for the vertical pass.
    v2f Bc[7];
#pragma unroll
    for (int ch = 0; ch < 7; ++ch) {
        int kk = 4 * ch + 2 * laneHi;
        int d0 = kk - laneLo, d1 = kk + 1 - laneLo;
        float b0 = (d0 >= 0 && d0 <= 10) ? gk[d0] : 0.0f;
        float b1 = (d1 >= 0 && d1 <= 10) ? gk[d1] : 0.0f;
        Bc[ch].x = b0; Bc[ch].y = b1;
    }

    // ---- load 26x26 halos of x and y (guarded at image edge) ----
    const int r0 = ty * TILE, c0 = tx * TILE;
    for (int idx = lane; idx < HALO * HALO; idx += 32) {
        int r = idx / HALO, c = idx - r * HALO;
        int gr = r0 + r, gc = c0 + c;
        bool ok = (gr < IMGHW) && (gc < IMGHW);
        size_t gi = (size_t)gr * IMGHW + gc;
        Xb[r * PSTR + c] = ok ? xim[gi] : 0.0f;
        Yb[r * PSTR + c] = ok ? yim[gi] : 0.0f;
    }
    // zero K-padding: P columns 26..27, H rows 26..27
    for (int idx = lane; idx < HALO * 2; idx += 32) {
        int r = idx >> 1, c = 26 + (idx & 1);
        Pb[r * PSTR + c] = 0.0f;
    }
    Hb[26 * 16 + lane] = 0.0f;             // 32 entries = rows 26,27
    __syncthreads();

    // ---- 5 blurred maps: x, y, x*x, y*y, x*y ----
    v8f res[5];
#pragma unroll
    for (int m = 0; m < 5; ++m) {
        // product map into Pb
        for (int idx = lane; idx < HALO * HALO; idx += 32) {
            int r = idx / HALO, c = idx - r * HALO;
            float xv = Xb[r * PSTR + c], yv = Yb[r * PSTR + c];
            float p;
            if      (m == 0) p = xv;
            else if (m == 1) p = yv;
            else if (m == 2) p = xv * xv;
            else if (m == 3) p = yv * yv;
            else             p = xv * yv;
            Pb[r * PSTR + c] = p;
        }
        __syncthreads();

        // horizontal: H[r][j] = sum_t gn[t] * P[r][j+t], rows 0..25 in 2 blocks
#pragma unroll
        for (int rb = 0; rb < 2; ++rb) {
            const int rowbase = rb ? 10 : 0;
            v8f acc = {0.f, 0.f, 0.f, 0.f, 0.f, 0.f, 0.f, 0.f};
#pragma unroll
            for (int ch = 0; ch < 7; ++ch) {
                int kk = 4 * ch + 2 * laneHi;
                v2f a = *(const v2f*)&Pb[(rowbase + laneLo) * PSTR + kk];
                acc = wmma4(a, Bc[ch], acc);
            }
#pragma unroll
            for (int v = 0; v < 8; ++v) {
                int row = rowbase + v + 8 * laneHi;      // global H row
                bool w = rb ? (row >= 16) : true;        // each row written once
                if (w) Hb[row * 16 + laneLo] = acc[v];
            }
        }
        __syncthreads();

        // vertical: Out[r][j] = sum_t gn[t] * H[r+t][j]
        {
            v8f acc = {0.f, 0.f, 0.f, 0.f, 0.f, 0.f, 0.f, 0.f};
#pragma unroll
            for (int ch = 0; ch < 7; ++ch) {
                int kk = 4 * ch + 2 * laneHi;
                v2f bb;
                bb.x = Hb[kk * 16 + laneLo];
                bb.y = Hb[(kk + 1) * 16 + laneLo];
                acc = wmma4(Bc[ch], bb, acc);
            }
            res[m] = acc;
        }
        __syncthreads();   // before next map overwrites Pb/Hb
    }

    // ---- SSIM map + masked partial sum ----
    const v8f mu1 = res[0], mu2 = res[1], bxx = res[2], byy = res[3], bxy = res[4];
    float lsum = 0.0f;
#pragma unroll
    for (int v = 0; v < 8; ++v) {
        int orow = r0 + v + 8 * laneHi;
        int ocol = c0 + laneLo;
        if (orow < OUTHW && ocol < OUTHW) {
            float m1 = mu1[v], m2 = mu2[v];
            float m1m2 = m1 * m2, m1s = m1 * m1, m2s = m2 * m2;
            float s1  = bxx[v] - m1s;
            float s2  = byy[v] - m2s;
            float s12 = bxy[v] - m1m2;
            float num = (2.0f * m1m2 + C1_SSIM) * (2.0f * s12 + C2_SSIM);
            float den = (m1s + m2s + C1_SSIM) * (s1 + s2 + C2_SSIM);
            lsum += num / den;
        }
    }
    // wave32 reduction
#pragma unroll
    for (int off = 16; off >= 1; off >>= 1)
        lsum += __shfl_xor(lsum, off, 32);
    if (lane == 0) partial[blockIdx.x] = lsum;
}

// Deterministic final reduction (single block, f64 accumulation).
__global__ void __launch_bounds__(256)
ssim_reduce_kernel(const float* __restrict__ partial, int n, float* __restrict__ out) {
    __shared__ double sd[256];
    double s = 0.0;
    for (int i = threadIdx.x; i < n; i += 256) s += (double)partial[i];
    sd[threadIdx.x] = s;
    __syncthreads();
    for (int off = 128; off >= 1; off >>= 1) {
        if ((int)threadIdx.x < off) sd[threadIdx.x] += sd[threadIdx.x + off];
        __syncthreads();
    }
    if (threadIdx.x == 0) out[0] = (float)(sd[0] * (1.0 / N_TOTAL));
}

extern "C" void kernel_launch(void* const* d_in, const int* in_sizes, int n_in,
                              void* d_out, int out_size, void* d_ws, size_t ws_size,
                              hipStream_t stream) {
    const float* X = (const float*)d_in[0];   // "output"
    const float* Y = (const float*)d_in[1];   // "target"
    float* out     = (float*)d_out;
    float* partial = (float*)d_ws;            // 98304 floats = 384 KB

    const int nTiles = 96 * NTILES1D * NTILES1D;   // 98304 waves / tiles
    ssim_tile_kernel<<<nTiles, 32, 0, stream>>>(X, Y, partial);
    ssim_reduce_kernel<<<1, 256, 0, stream>>>(partial, nTiles, out);
}